// SparseMax_66769561583800
// MI455X (gfx1250) — compile-verified
//
#include <hip/hip_runtime.h>

#define D      4096
#define TPB    256     // 8 waves of 32
#define ELEMS  16      // columns per thread (TPB * ELEMS == D)

typedef __attribute__((ext_vector_type(2))) float v2f;
typedef __attribute__((ext_vector_type(8))) float v8f;

__device__ __forceinline__ float wave_sum(float v) {
#pragma unroll
    for (int off = 16; off > 0; off >>= 1) v += __shfl_xor(v, off, 32);
    return v;
}

// ---------------------------------------------------------------------------
// Kernel A: per-row sparsemax tau via Michelot fixed-point iteration
//           (exact, sort-free), accumulate relu(z - tau) into per-workgroup
//           partial column sums. Single streaming pass over z.
// ---------------------------------------------------------------------------
__global__ void sparsemax_tau_partial(const float* __restrict__ z,
                                      float* __restrict__ partial,
                                      int nrows, int ngroups) {
    __shared__ float red_s[8];
    __shared__ float red_c[8];
    __shared__ float tau_sh;

    const int t    = threadIdx.x;
    const int wid  = t >> 5;
    const int lane = t & 31;

    float acc[ELEMS];
#pragma unroll
    for (int j = 0; j < ELEMS; ++j) acc[j] = 0.f;

    for (int row = blockIdx.x; row < nrows; row += ngroups) {
        // --- load this thread's 16 columns (coalesced 4x b128) ---
        float x[ELEMS];
        {
            const float4* rp4 =
                reinterpret_cast<const float4*>(z + (size_t)row * D + t * ELEMS);
            float4* xv = reinterpret_cast<float4*>(x);
            xv[0] = rp4[0]; xv[1] = rp4[1]; xv[2] = rp4[2]; xv[3] = rp4[3];
        }
        // gfx1250 prefetch of next row owned by this workgroup
        if (row + ngroups < nrows)
            __builtin_prefetch(z + (size_t)(row + ngroups) * D + t * ELEMS, 0, 1);

        // --- Michelot init: tau0 = (sum(all) - 1) / D ---
        float tau;
        {
            float s = 0.f;
#pragma unroll
            for (int j = 0; j < ELEMS; ++j) s += x[j];
            s = wave_sum(s);
            if (lane == 0) red_s[wid] = s;
            __syncthreads();
            if (wid == 0) {
                float S = (lane < 8) ? red_s[lane] : 0.f;
                S = wave_sum(S);
                if (lane == 0) tau_sh = (S - 1.f) * (1.f / (float)D);
            }
            __syncthreads();
            tau = tau_sh;
        }

        // --- fixed-point iterations: tau <- (sum_{z>tau} z - 1)/count ---
#pragma unroll 1
        for (int it = 0; it < 32; ++it) {
            float s = 0.f, c = 0.f;
#pragma unroll
            for (int j = 0; j < ELEMS; ++j) {
                bool g = x[j] > tau;
                s += g ? x[j] : 0.f;
                c += g ? 1.f : 0.f;
            }
            s = wave_sum(s);
            c = wave_sum(c);
            if (lane == 0) { red_s[wid] = s; red_c[wid] = c; }
            __syncthreads();
            if (wid == 0) {
                float S = (lane < 8) ? red_s[lane] : 0.f;
                float C = (lane < 8) ? red_c[lane] : 0.f;
                S = wave_sum(S);
                C = wave_sum(C);
                if (lane == 0) tau_sh = (S - 1.f) / C;
            }
            __syncthreads();
            float newtau = tau_sh;           // block-uniform
            if (newtau == tau) break;        // converged fixed point (uniform branch)
            tau = newtau;
        }

        // --- accumulate relu(z - tau) for this workgroup's column partials ---
#pragma unroll
        for (int j = 0; j < ELEMS; ++j)
            acc[j] += fmaxf(x[j] - tau, 0.f);
        __syncthreads();  // protect shared arrays before next row's reduction
    }

    // deterministic per-workgroup partial sums (no float atomics)
    float4* pp = reinterpret_cast<float4*>(partial + (size_t)blockIdx.x * D + t * ELEMS);
    const float4* av = reinterpret_cast<const float4*>(acc);
    pp[0] = av[0]; pp[1] = av[1]; pp[2] = av[2]; pp[3] = av[3];
}

// ---------------------------------------------------------------------------
// Kernel B: column-sum of partial[W x 4096] via WMMA ones-GEMV:
//           C += ones(16x4) * B(4x16)  — every row of C holds the column sums.
//           One wave handles 16 columns. A = ones, so the K-row <-> VGPR-half
//           mapping is irrelevant; only the N <-> lane mapping matters.
// ---------------------------------------------------------------------------
__global__ void colsum_wmma(const float* __restrict__ partial,
                            float* __restrict__ p,
                            int W, float invN) {
    const int lane = threadIdx.x & 31;
    const int gwave = (blockIdx.x * blockDim.x + threadIdx.x) >> 5;
    const int base = gwave * 16;             // column tile base
    if (base >= D) return;                   // wave-uniform guard

    v8f c = {0.f, 0.f, 0.f, 0.f, 0.f, 0.f, 0.f, 0.f};
    v2f a; a.x = 1.f; a.y = 1.f;             // all-ones A (16x4 f32 -> 2 VGPRs)

    const int n    = lane & 15;              // N index within tile
    const int half = lane >> 4;

    for (int k = 0; k < W; k += 4) {
        v2f b;
        b.x = partial[(size_t)(k + half)     * D + base + n];   // rows k, k+1
        b.y = partial[(size_t)(k + 2 + half) * D + base + n];   // rows k+2, k+3
        c = __builtin_amdgcn_wmma_f32_16x16x4_f32(
                /*neg_a=*/false, a, /*neg_b=*/false, b,
                /*c_mod=*/(short)0, c, /*reuse_a=*/false, /*reuse_b=*/false);
    }

    // C row M=0 lives in VGPR0 of lanes 0..15 (N = lane); all rows identical.
    if (lane < 16) p[base + lane] = c[0] * invN;
}

// ---------------------------------------------------------------------------
// Kernel C: broadcast p[4096] to all rows (coalesced float4 stores, 256 MB).
// ---------------------------------------------------------------------------
__global__ void broadcast_rows(const float* __restrict__ p,
                               float4* __restrict__ out, int nrows) {
    const int gid = blockIdx.x * blockDim.x + threadIdx.x;
    const int cq = gid & (D / 4 - 1);                  // column quad 0..1023
    const int rstart = gid >> 10;
    const int rstride = (gridDim.x * blockDim.x) >> 10;
    const float4 v = reinterpret_cast<const float4*>(p)[cq];
    for (int r = rstart; r < nrows; r += rstride)
        out[(size_t)r * (D / 4) + cq] = v;
}

extern "C" void kernel_launch(void* const* d_in, const int* in_sizes, int n_in,
                              void* d_out, int out_size, void* d_ws, size_t ws_size,
                              hipStream_t stream) {
    const float* z = (const float*)d_in[0];
    float* out = (float*)d_out;
    const int nrows = in_sizes[0] / D;

    float* ws = (float*)d_ws;
    float* p = ws;                 // [4096] final mean vector
    float* partial = ws + D;       // [W][4096] per-workgroup partials

    // choose W (number of workgroups in kernel A) from workspace budget
    long maxW = ((long)(ws_size / sizeof(float)) - D) / D;
    int W = 512;
    if ((long)W > maxW) W = (int)maxW;
    W &= ~3;                       // multiple of 4 for the WMMA K-loop
    if (W < 4) W = 4;

    sparsemax_tau_partial<<<W, TPB, 0, stream>>>(z, partial, nrows, W);

    // D/16 = 256 waves -> 8192 threads -> 32 blocks of 256
    colsum_wmma<<<(D / 16 * 32 + TPB - 1) / TPB, TPB, 0, stream>>>(
        partial, p, W, 1.f / (float)nrows);

    broadcast_rows<<<4096, TPB, 0, stream>>>(p, (float4*)out, nrows);
}